// UnsupervisedECAModel_45707041964808
// MI455X (gfx1250) — compile-verified
//
#include <hip/hip_runtime.h>

// ---------------------------------------------------------------------------
// UnsupervisedECAModel for MI455X (gfx1250, wave32, WMMA)
//
//   A = A_raw - A_raw^T + diag(D)
//   P = expm(A)  via scaling-and-squaring + 18-term Taylor (Horner), all fp32
//   L = sigmoid(L_raw); PL = P @ L; proj = X @ PL; probs = softmax(proj)
//   outputs (flat, in order): probs | prob(=proj) | P | proj
//
// All GEMMs use V_WMMA_F32_16X16X4_F32 (fp32 matrix path on CDNA5).
// B tiles staged transposed in LDS (single ds_load_b64 per fragment), and the
// global->LDS path is software-pipelined with double-buffered LDS so global
// load latency overlaps the WMMA block instead of stalling before it.
// ---------------------------------------------------------------------------

typedef float v2f __attribute__((ext_vector_type(2)));
typedef float v8f __attribute__((ext_vector_type(8)));

#define NSAMP 65536
#define DIM   1024
#define NC    64

#define BK     32
#define LDS_ST 36   // LDS row stride (floats): 16B-aligned, conflict-free frag reads

// ---- A = A_raw - A_raw^T + diag(D);  T = I + alpha*A (innermost Horner term)
__global__ __launch_bounds__(256) void build_A_kernel(
    const float* __restrict__ A_raw, const float* __restrict__ D,
    float* __restrict__ A, float* __restrict__ T, float alpha)
{
    int idx = blockIdx.x * 256 + threadIdx.x;
    int i = idx >> 10, j = idx & (DIM - 1);
    float a = A_raw[idx] - A_raw[j * DIM + i];
    if (i == j) a += D[i];
    A[idx] = a;
    T[idx] = ((i == j) ? 1.0f : 0.0f) + alpha * a;
}

__global__ __launch_bounds__(256) void sigmoid_kernel(
    const float* __restrict__ in, float* __restrict__ out)
{
    int idx = blockIdx.x * 256 + threadIdx.x;
    out[idx] = 1.0f / (1.0f + __expf(-in[idx]));
}

// ---- Generic fp32 WMMA GEMM:  C = alpha * (A @ B) [+ I]
// A: MxK row-major, B: KxN row-major, C: MxN row-major.
// Block = 256 threads = 8 waves arranged WAVES_M x WAVES_N; each wave owns an
// (MT*16) x (NT*16) tile of 16x16 WMMA accumulators.
// M % (WAVES_M*MT*16) == 0, N % (WAVES_N*NT*16) == 0, K % 32 == 0.
template <int WAVES_M, int WAVES_N, int MT, int NT>
__global__ __launch_bounds__(256) void gemm_wmma_f32(
    const float* __restrict__ A, const float* __restrict__ B,
    float* __restrict__ C, int M, int N, int K, float alpha, int addIdentity)
{
    constexpr int BMt = WAVES_M * MT * 16;
    constexpr int BNt = WAVES_N * NT * 16;
    constexpr int A_ITERS = BMt * BK / (4 * 256);
    constexpr int B_ITERS = BK * BNt / (4 * 256);

    __shared__ float As[2][BMt * LDS_ST];   // [buf][row][k]
    __shared__ float Bt[2][BNt * LDS_ST];   // [buf][col][k]  (transposed B tile)

    const int tid      = threadIdx.x;
    const int lane     = tid & 31;
    const int wid      = tid >> 5;
    const int wm       = (wid / WAVES_N) * (MT * 16);
    const int wn       = (wid % WAVES_N) * (NT * 16);
    const int laneHalf = lane >> 4;         // WMMA fragment K-half select
    const int lane15   = lane & 15;

    const int blockM = blockIdx.y * BMt;
    const int blockN = blockIdx.x * BNt;

    float4 aReg[A_ITERS];
    float4 bReg[B_ITERS];

    auto loadRegs = [&](int k0) {
        #pragma unroll
        for (int it = 0; it < A_ITERS; ++it) {
            int idx = it * 256 + tid;
            int row = idx / (BK / 4);
            int c4  = (idx % (BK / 4)) * 4;
            aReg[it] = *reinterpret_cast<const float4*>(
                &A[(size_t)(blockM + row) * K + k0 + c4]);
        }
        #pragma unroll
        for (int it = 0; it < B_ITERS; ++it) {
            int idx = it * 256 + tid;
            int row = idx / (BNt / 4);          // k within chunk
            int c4  = (idx % (BNt / 4)) * 4;    // col within tile
            bReg[it] = *reinterpret_cast<const float4*>(
                &B[(size_t)(k0 + row) * N + blockN + c4]);
        }
    };
    auto storeLDS = [&](int buf) {
        #pragma unroll
        for (int it = 0; it < A_ITERS; ++it) {
            int idx = it * 256 + tid;
            int row = idx / (BK / 4);
            int c4  = (idx % (BK / 4)) * 4;
            *reinterpret_cast<float4*>(&As[buf][row * LDS_ST + c4]) = aReg[it];
        }
        #pragma unroll
        for (int it = 0; it < B_ITERS; ++it) {
            int idx = it * 256 + tid;
            int row = idx / (BNt / 4);
            int c4  = (idx % (BNt / 4)) * 4;
            float4 v = bReg[it];
            Bt[buf][(c4 + 0) * LDS_ST + row] = v.x;
            Bt[buf][(c4 + 1) * LDS_ST + row] = v.y;
            Bt[buf][(c4 + 2) * LDS_ST + row] = v.z;
            Bt[buf][(c4 + 3) * LDS_ST + row] = v.w;
        }
    };

    v8f acc[MT][NT];
    #pragma unroll
    for (int mi = 0; mi < MT; ++mi)
        #pragma unroll
        for (int ni = 0; ni < NT; ++ni)
            acc[mi][ni] = (v8f){0.f, 0.f, 0.f, 0.f, 0.f, 0.f, 0.f, 0.f};

    // Pipeline prologue: chunk 0 -> LDS buffer 0
    loadRegs(0);
    storeLDS(0);
    __syncthreads();

    int buf = 0;
    for (int k0 = 0; k0 < K; k0 += BK) {
        const bool hasNext = (k0 + BK) < K;
        if (hasNext) loadRegs(k0 + BK);   // issue global loads; wait lands in storeLDS

        #pragma unroll
        for (int kk = 0; kk < BK; kk += 4) {
            const int kc = kk + laneHalf * 2;
            // A fragments (16x4): lanes 0-15 -> K=kk..kk+1, lanes 16-31 -> kk+2..kk+3
            v2f afrag[MT];
            #pragma unroll
            for (int mi = 0; mi < MT; ++mi)
                afrag[mi] = *reinterpret_cast<const v2f*>(
                    &As[buf][(wm + mi * 16 + lane15) * LDS_ST + kc]);
            // B fragments (4x16): contiguous in transposed layout -> ds_load_b64
            v2f bfrag[NT];
            #pragma unroll
            for (int ni = 0; ni < NT; ++ni)
                bfrag[ni] = *reinterpret_cast<const v2f*>(
                    &Bt[buf][(wn + ni * 16 + lane15) * LDS_ST + kc]);
            #pragma unroll
            for (int mi = 0; mi < MT; ++mi)
                #pragma unroll
                for (int ni = 0; ni < NT; ++ni)
                    acc[mi][ni] = __builtin_amdgcn_wmma_f32_16x16x4_f32(
                        false, afrag[mi], false, bfrag[ni],
                        (short)0, acc[mi][ni], false, false);
        }

        if (hasNext) {
            storeLDS(buf ^ 1);
            __syncthreads();
            buf ^= 1;
        }
    }

    // Epilogue: C = alpha*acc (+I on diagonal). C/D layout: VGPR r holds
    // M = r (lanes 0-15) / r+8 (lanes 16-31), N = lane%16.
    #pragma unroll
    for (int mi = 0; mi < MT; ++mi) {
        #pragma unroll
        for (int ni = 0; ni < NT; ++ni) {
            int gm0 = blockM + wm + mi * 16 + laneHalf * 8;
            int gn  = blockN + wn + ni * 16 + lane15;
            #pragma unroll
            for (int r = 0; r < 8; ++r) {
                int gm  = gm0 + r;
                float v = alpha * acc[mi][ni][r];
                if (addIdentity && (gm == gn)) v += 1.0f;
                C[(size_t)gm * N + gn] = v;
            }
        }
    }
}

// ---- softmax over 64 columns, one wave32 per row; also writes prob = proj copy
__global__ __launch_bounds__(256) void softmax_copy_kernel(
    const float* __restrict__ proj, float* __restrict__ probs,
    float* __restrict__ prob)
{
    int lane = threadIdx.x & 31;
    int wid  = threadIdx.x >> 5;
    size_t row = (size_t)blockIdx.x * 8 + wid;
    const float* p = proj + row * NC;
    float v0 = p[lane], v1 = p[lane + 32];
    float m = fmaxf(v0, v1);
    #pragma unroll
    for (int off = 16; off > 0; off >>= 1)
        m = fmaxf(m, __shfl_xor(m, off, 32));
    float e0 = __expf(v0 - m), e1 = __expf(v1 - m);
    float s = e0 + e1;
    #pragma unroll
    for (int off = 16; off > 0; off >>= 1)
        s += __shfl_xor(s, off, 32);
    float inv = 1.0f / s;
    probs[row * NC + lane]      = e0 * inv;
    probs[row * NC + lane + 32] = e1 * inv;
    prob[row * NC + lane]       = v0;
    prob[row * NC + lane + 32]  = v1;
}

extern "C" void kernel_launch(void* const* d_in, const int* in_sizes, int n_in,
                              void* d_out, int out_size, void* d_ws, size_t ws_size,
                              hipStream_t stream)
{
    (void)in_sizes; (void)n_in; (void)out_size; (void)ws_size;

    const float* X     = (const float*)d_in[0];   // 65536 x 1024
    const float* A_raw = (const float*)d_in[1];   // 1024 x 1024
    const float* Dvec  = (const float*)d_in[2];   // 1024
    const float* L_raw = (const float*)d_in[3];   // 1024 x 64

    float* out   = (float*)d_out;
    float* probs = out;                                   // 65536*64
    float* prob  = probs + (size_t)NSAMP * NC;            // 65536*64
    float* Pout  = prob  + (size_t)NSAMP * NC;            // 1024*1024 (also ping-pong buf)
    float* proj  = Pout  + (size_t)DIM * DIM;             // 65536*64

    float* w    = (float*)d_ws;
    float* Amat = w;                         // 1024*1024
    float* T1   = Amat + (size_t)DIM * DIM;  // 1024*1024 (ping-pong partner of Pout)
    float* Ls   = T1   + (size_t)DIM * DIM;  // 1024*64
    float* PL   = Ls   + (size_t)DIM * NC;   // 1024*64
    // total workspace: ~8.5 MB

    const int   S_SQ  = 10;                       // squarings
    const int   M_TAY = 18;                       // Taylor degree
    const float scale = 1.0f / (float)(1 << S_SQ);   // 2^-10; folded into GEMM alpha

    // A and innermost Horner term T_18 = I + (scale/18)*A
    build_A_kernel<<<DIM * DIM / 256, 256, 0, stream>>>(
        A_raw, Dvec, Amat, T1, scale / (float)M_TAY);

    // Big square GEMMs: 128x128 block tile, wave tile 64x32 (8 WMMA / 6 frag loads)
    dim3 gridSq(DIM / 128, DIM / 128);

    // Horner: T_j = I + (scale/j) * (A @ T_{j+1}), j = 17..1  (17 GEMMs)
    float* src = T1; float* dst = Pout;
    for (int j = M_TAY - 1; j >= 1; --j) {
        gemm_wmma_f32<2, 4, 4, 2><<<gridSq, 256, 0, stream>>>(
            Amat, src, dst, DIM, DIM, DIM, scale / (float)j, 1);
        float* t = src; src = dst; dst = t;
    }
    // 17 GEMMs (odd) -> T_1 now in Pout, src == Pout.

    // Squaring: P <- P @ P, 10 times (even) -> final result lands back in Pout.
    for (int sq = 0; sq < S_SQ; ++sq) {
        gemm_wmma_f32<2, 4, 4, 2><<<gridSq, 256, 0, stream>>>(
            src, src, dst, DIM, DIM, DIM, 1.0f, 0);
        float* t = src; src = dst; dst = t;
    }

    // L = sigmoid(L_raw); PL = P @ L; proj = X @ PL  (N=64 config: 128x64 tiles)
    sigmoid_kernel<<<DIM * NC / 256, 256, 0, stream>>>(L_raw, Ls);
    gemm_wmma_f32<4, 2, 2, 2><<<dim3(1, DIM / 128), 256, 0, stream>>>(
        Pout, Ls, PL, DIM, NC, DIM, 1.0f, 0);
    gemm_wmma_f32<4, 2, 2, 2><<<dim3(1, NSAMP / 128), 256, 0, stream>>>(
        X, PL, proj, NSAMP, NC, DIM, 1.0f, 0);

    // probs = softmax(proj); prob = proj (copy)
    softmax_copy_kernel<<<NSAMP / 8, 256, 0, stream>>>(proj, probs, prob);
}